// SplineGCN_13967233647446
// MI455X (gfx1250) — compile-verified
//
#include <hip/hip_runtime.h>
#include <hip/hip_bf16.h>

// Problem constants (match reference)
#define N_NODES 100000
#define N_EDGES 1600000
#define IN_F    32
#define OUT_F   64
#define KS      5
#define K_TOTAL 25          // 5*5
#define KDIM    800         // K_TOTAL * IN_F
#define H_FLOATS ((long)N_NODES * K_TOTAL * IN_F)   // 80,000,000 floats = 320 MB

typedef __attribute__((ext_vector_type(2))) float v2f;
typedef __attribute__((ext_vector_type(8))) float v8f;
typedef __attribute__((ext_vector_type(4))) unsigned int u32x4;
typedef __attribute__((ext_vector_type(8))) int i32x8;
typedef __attribute__((ext_vector_type(4))) int i32x4;

// ---------------------------------------------------------------------------
// TDM descriptor builder + issue (CDNA5 Tensor Data Mover, 2D tile -> LDS).
// D# packing per CDNA5 ISA ch.8.3/8.4:
//   group0: [1:0]=count, [63:32]=lds_addr, [120:64]=global tile addr, [127:126]=type(2)
//   group1: [17:16]=data_size, [20]=pad_en, [24:22]=pad_interval, [31:25]=pad_amount,
//           [79:48]=tensor_dim0, [111:80]=tensor_dim1, [127:112]=tile_dim0,
//           [143:128]=tile_dim1, [159:144]=tile_dim2, [207:160]=tensor_dim0_stride
//   groups 2/3: zero (tensor rank <= 2)
// This toolchain exposes the 6-arg builtin form (extra int32x8 before cpol).
// ---------------------------------------------------------------------------
__device__ __forceinline__ void tdm_load_2d_to_lds(unsigned lds_addr,
                                                   const void* gaddr,
                                                   unsigned tensor_d0,
                                                   unsigned tensor_d1,
                                                   unsigned tile_d0,
                                                   unsigned tile_d1,
                                                   unsigned stride0,
                                                   bool pad_row) {
    const unsigned long long ga = (unsigned long long)(size_t)gaddr;
    u32x4 g0;
    g0[0] = 1u;                                              // count=1, user desc
    g0[1] = lds_addr;                                        // LDS byte address
    g0[2] = (unsigned)(ga & 0xFFFFFFFFull);                  // global addr [31:0]
    g0[3] = (unsigned)((ga >> 32) & 0x01FFFFFFull)           // global addr [56:32]
          | (2u << 30);                                      // type = 2 ("image")

    unsigned w0 = (2u << 16);                                // data_size = 4 bytes
    if (pad_row) w0 |= (1u << 20)                            // pad_enable
               | (4u << 22)                                  // pad_interval: 32 DWORDs
               | (0u << 25);                                 // pad_amount: 1 DWORD
    i32x8 g1;
    g1[0] = (int)w0;
    g1[1] = (int)((tensor_d0 & 0xFFFFu) << 16);              // tensor_dim0[15:0]
    g1[2] = (int)((tensor_d0 >> 16) | ((tensor_d1 & 0xFFFFu) << 16));
    g1[3] = (int)((tensor_d1 >> 16) | (tile_d0 << 16));      // tile_dim0
    g1[4] = (int)tile_d1;                                    // tile_dim1 (tile_dim2=0)
    g1[5] = (int)stride0;                                    // dim0 stride [31:0]
    g1[6] = 0;
    g1[7] = 0;
    const i32x4 z4 = {0, 0, 0, 0};
    const i32x8 z8 = {0, 0, 0, 0, 0, 0, 0, 0};
    __builtin_amdgcn_tensor_load_to_lds(g0, g1, z4, z4, z8, 0);
}

// ---------------------------------------------------------------------------
// Kernel 1: zero the H accumulator (320 MB) with wide stores
// ---------------------------------------------------------------------------
__global__ __launch_bounds__(256)
void spline_zero_kernel(float4* __restrict__ p, long n4) {
    long i = (long)blockIdx.x * blockDim.x + threadIdx.x;
    if (i < n4) {
        p[i] = make_float4(0.f, 0.f, 0.f, 0.f);
    }
}

// ---------------------------------------------------------------------------
// Kernel 2: per-edge bilinear spline scatter into H.
// One wave (32 lanes) per edge; lane == input-feature index (IN_F == 32,
// wave32 on gfx1250). 4 native f32 global atomics per lane per edge.
// ---------------------------------------------------------------------------
__global__ __launch_bounds__(256)
void spline_scatter_kernel(const float* __restrict__ feat,
                           const float* __restrict__ pseudo,
                           const int*   __restrict__ edge_index,
                           float*       __restrict__ H) {
    const int tid  = threadIdx.x;
    const int lane = tid & 31;
    const long e   = (long)blockIdx.x * 8 + (tid >> 5);
    if (e >= N_EDGES) return;

    const int row = edge_index[e];            // dest
    const int col = edge_index[N_EDGES + e];  // src

    // degree-1 open B-spline, kernel size 5 per dim -> v in [0,4]
    const float p0 = pseudo[2 * e + 0];
    const float p1 = pseudo[2 * e + 1];
    const float v0 = p0 * 4.0f;
    const float v1 = p1 * 4.0f;
    int i0 = (int)v0; i0 = i0 < 0 ? 0 : (i0 > 3 ? 3 : i0);
    int i1 = (int)v1; i1 = i1 < 0 ? 0 : (i1 > 3 ? 3 : i1);
    const float f0 = v0 - (float)i0;
    const float f1 = v1 - (float)i1;

    // corner weights / flat kernel bins (strides [5,1], combo order 00,01,10,11)
    const float w00 = (1.f - f0) * (1.f - f1);
    const float w01 = (1.f - f0) * f1;
    const float w10 = f0 * (1.f - f1);
    const float w11 = f0 * f1;
    const int k00 = i0 * KS + i1;
    const int k01 = i0 * KS + i1 + 1;
    const int k10 = (i0 + 1) * KS + i1;
    const int k11 = (i0 + 1) * KS + i1 + 1;

    const float xv = feat[(long)col * IN_F + lane];
    float* __restrict__ Hrow = H + ((long)row * K_TOTAL) * IN_F + lane;

    unsafeAtomicAdd(Hrow + (long)k00 * IN_F, w00 * xv);
    unsafeAtomicAdd(Hrow + (long)k01 * IN_F, w01 * xv);
    unsafeAtomicAdd(Hrow + (long)k10 * IN_F, w10 * xv);
    unsafeAtomicAdd(Hrow + (long)k11 * IN_F, w11 * xv);
}

// ---------------------------------------------------------------------------
// Kernel 3: out[100000,64] = H2[100000,800] @ Wr[800,64] + bias
// Full f32 precision via V_WMMA_F32_16X16X4_F32.
// Block = 256 threads = 8 waves -> 32x64 output tile (each wave a 16x16 tile).
// Tiles staged HBM->LDS by the Tensor Data Mover (double-buffered; the TDM
// transfer of step k+1 overlaps the WMMAs of step k). The A-tile descriptor
// uses TDM row padding (1 DWORD per 32) to realize the bank-conflict-free
// [32][33] LDS layout in hardware.
// ---------------------------------------------------------------------------
__global__ __launch_bounds__(256)
void spline_gemm_kernel(const float* __restrict__ H,
                        const float* __restrict__ Wr,
                        const float* __restrict__ bias,
                        float*       __restrict__ out) {
    __shared__ float As[2][32][33];   // [buf][m][k] tile of H2 (TDM-padded rows)
    __shared__ float Bs[2][32][64];   // [buf][k][n] tile of Wr (contiguous chunk)

    const int tid  = threadIdx.x;
    const int wave = tid >> 5;
    const int lane = tid & 31;
    const int m_off = (wave >> 2) << 4;   // 0 / 16
    const int n_off = (wave & 3) << 4;    // 0 / 16 / 32 / 48
    const long row_base = (long)blockIdx.x * 32;

    const int ml    = lane & 15;          // M (for A) or N (for B/C/D) index
    const int khalf = (lane >> 4) << 1;   // 0 or 2: K sub-pair selected by lane half

    v8f acc = {0.f, 0.f, 0.f, 0.f, 0.f, 0.f, 0.f, 0.f};

    // ---- TDM tile issue: one wave only (TDM ignores EXEC; avoid 8x duplicates)
    auto issue_tiles = [&](int kk, int buf) {
        const int kbase = kk * 32;
        // A: 32 rows x 32 k from H2[row_base.., kbase..], row stride KDIM,
        //    padded in LDS to 33 floats/row.
        tdm_load_2d_to_lds((unsigned)(size_t)&As[buf][0][0],
                           H + row_base * (long)KDIM + kbase,
                           /*tensor_d0=*/KDIM, /*tensor_d1=*/N_NODES,
                           /*tile_d0=*/32, /*tile_d1=*/32,
                           /*stride0=*/KDIM, /*pad_row=*/true);
        // B: Wr rows kbase..kbase+31 are one dense 2048-float chunk -> 1D tile.
        tdm_load_2d_to_lds((unsigned)(size_t)&Bs[buf][0][0],
                           Wr + (long)kbase * OUT_F,
                           /*tensor_d0=*/KDIM * OUT_F, /*tensor_d1=*/1,
                           /*tile_d0=*/2048, /*tile_d1=*/0,
                           /*stride0=*/2048, /*pad_row=*/false);
    };

    if (wave == 0) {
        issue_tiles(0, 0);
    }

    for (int kk = 0; kk < KDIM / 32; ++kk) {     // 25 K-steps
        const int cur = kk & 1;
        if (wave == 0) {
            __builtin_amdgcn_s_wait_tensorcnt(0);    // tile kk landed in LDS
        }
        __syncthreads();                             // publish LDS to all waves
        if (wave == 0 && kk + 1 < KDIM / 32) {
            issue_tiles(kk + 1, cur ^ 1);            // overlap next DMA w/ compute
        }

        // ---- 8 WMMAs of K=4 each from LDS fragments
        #pragma unroll
        for (int k4 = 0; k4 < 8; ++k4) {
            const int kb = k4 * 4 + khalf;
            v2f a, b;
            // A 16x4 f32 layout: lane m=L%16; VGPR0 holds K=khalf, VGPR1 K=khalf+1
            a.x = As[cur][m_off + ml][kb];
            a.y = As[cur][m_off + ml][kb + 1];
            // B 4x16 f32 layout: lane n=L%16; VGPR0 holds K=khalf, VGPR1 K=khalf+1
            b.x = Bs[cur][kb][n_off + ml];
            b.y = Bs[cur][kb + 1][n_off + ml];
            acc = __builtin_amdgcn_wmma_f32_16x16x4_f32(
                /*neg_a=*/false, a, /*neg_b=*/false, b,
                /*c_mod=*/(short)0, acc, /*reuse_a=*/false, /*reuse_b=*/false);
        }
    }

    // ---- writeback with bias; C/D layout: VGPR r <-> M = r (+8 for high half)
    const float bv = bias[n_off + ml];
    const int mtop = (lane >> 4) << 3;           // 0 or 8
    #pragma unroll
    for (int r = 0; r < 8; ++r) {
        const long m = row_base + m_off + mtop + r;
        out[m * OUT_F + n_off + ml] = acc[r] + bv;
    }
}

// ---------------------------------------------------------------------------
// Host-side launcher
// Inputs (setup_inputs order):
//   d_in[0] features  [100000,32]  f32
//   d_in[1] pseudo    [1600000,2]  f32
//   d_in[2] edge_index[2,1600000]  i32
//   d_in[3] weight    [5,5,32,64]  f32  (flat == Wr[800][64])
//   d_in[4] bias      [64]         f32
// d_out: [100000,64] f32.  d_ws: H accumulator (needs 320 MB).
// ---------------------------------------------------------------------------
extern "C" void kernel_launch(void* const* d_in, const int* in_sizes, int n_in,
                              void* d_out, int out_size, void* d_ws, size_t ws_size,
                              hipStream_t stream) {
    const float* feat   = (const float*)d_in[0];
    const float* pseudo = (const float*)d_in[1];
    const int*   eidx   = (const int*)  d_in[2];
    const float* weight = (const float*)d_in[3];
    const float* bias   = (const float*)d_in[4];
    float* out = (float*)d_out;
    float* H   = (float*)d_ws;

    // 1) zero H (80M floats -> 20M float4)
    {
        const long n4 = H_FLOATS / 4;
        const int  thr = 256;
        const long blk = (n4 + thr - 1) / thr;
        spline_zero_kernel<<<(dim3)(unsigned)blk, thr, 0, stream>>>((float4*)H, n4);
    }
    // 2) scatter (8 edges per 256-thread block)
    {
        const int thr = 256;
        const long blk = (long)(N_EDGES + 7) / 8;
        spline_scatter_kernel<<<(dim3)(unsigned)blk, thr, 0, stream>>>(feat, pseudo, eidx, H);
    }
    // 3) WMMA GEMM + bias (32 rows per block; 100000 / 32 = 3125 exactly)
    {
        const int thr = 256;
        const int blk = N_NODES / 32;
        spline_gemm_kernel<<<blk, thr, 0, stream>>>(H, weight, bias, out);
    }
}